// DeltaNet_45397804319521
// MI455X (gfx1250) — compile-verified
//
#include <hip/hip_runtime.h>
#include <math.h>
#include <stdint.h>

// Problem constants (match reference)
#define Hdim 512
#define Bsz  4
#define Tlen 512
#define CKW  4            // depthwise conv taps
#define CH   16           // chunk length for WY delta-rule
#define NC   (Tlen / CH)  // 32 chunks
#define BT   (Bsz * Tlen) // 2048 rows
#define BTH  (BT * Hdim)  // 1,048,576 elements
#define SROW 516          // padded LDS row stride (floats) for staged 16x512 tiles

typedef float v2f __attribute__((ext_vector_type(2)));
typedef float v8f __attribute__((ext_vector_type(8)));

__device__ __forceinline__ float siluf(float x) { return x / (1.f + __expf(-x)); }
__device__ __forceinline__ float sigmf(float x) { return 1.f / (1.f + __expf(-x)); }

// Async copy of 16 bytes global -> LDS (ASYNCcnt-tracked, gfx1250)
__device__ __forceinline__ void async_b128(const float* gp, float* lp) {
  uint64_t ga = (uint64_t)(uintptr_t)gp;
  uint32_t la = (uint32_t)(uintptr_t)lp;   // low 32 bits of generic ptr = LDS offset
  asm volatile("global_load_async_to_lds_b128 %0, %1, off" :: "v"(la), "v"(ga) : "memory");
}
__device__ __forceinline__ void wait_async0() {
  asm volatile("s_wait_asynccnt 0x0" ::: "memory");
}

// ---------------------------------------------------------------------------
// GEMM: Out[r, n] = sum_c X[r, c] * W[n, c]      (X:[BT,H], W:[H,H], row-major)
// One wave per 16x16 output tile, fp32 WMMA, K stepped by 4.
// ---------------------------------------------------------------------------
__global__ void k_gemm_xWT(const float* __restrict__ X, const float* __restrict__ W,
                           float* __restrict__ Out) {
  const int lane = threadIdx.x & 31, hl = lane >> 4, idx = lane & 15;
  const float* A  = X + (size_t)blockIdx.x * 16 * Hdim;
  const float* Bp = W + (size_t)blockIdx.y * 16 * Hdim;
  v8f acc = {};
  for (int kk = 0; kk < Hdim / 4; ++kk) {
    const int k0 = kk * 4 + hl * 2;
    v2f a, b;
    a.x = A[idx * Hdim + k0];      a.y = A[idx * Hdim + k0 + 1];
    b.x = Bp[idx * Hdim + k0];     b.y = Bp[idx * Hdim + k0 + 1];
    acc = __builtin_amdgcn_wmma_f32_16x16x4_f32(false, a, false, b, (short)0, acc,
                                                false, false);
  }
  float* O = Out + (size_t)blockIdx.x * 16 * Hdim + blockIdx.y * 16;
#pragma unroll
  for (int v = 0; v < 8; ++v) O[(v + 8 * hl) * Hdim + idx] = acc[v];
}

// beta[r] = sigmoid(dot(x[r,:], Wb))
__global__ void k_beta(const float* __restrict__ x, const float* __restrict__ Wb,
                       float* __restrict__ beta) {
  int r = blockIdx.x * blockDim.x + threadIdx.x;
  if (r >= BT) return;
  float s = 0.f;
  for (int c = 0; c < Hdim; ++c) s += x[(size_t)r * Hdim + c] * Wb[c];
  beta[r] = sigmf(s);
}

// causal depthwise conv (K=4) + double SiLU, elementwise over [B,T,H]
__global__ void k_conv_silu(const float* __restrict__ raw, const float* __restrict__ w,
                            float* __restrict__ out) {
  size_t i = (size_t)blockIdx.x * blockDim.x + threadIdx.x;
  if (i >= (size_t)BTH) return;
  int h = (int)(i % Hdim);
  size_t bt = i / Hdim;
  int t = (int)(bt % Tlen);
  float acc = 0.f;
#pragma unroll
  for (int j = 0; j < CKW; ++j) {
    int ts = t - (CKW - 1) + j;
    if (ts >= 0) acc += raw[i + (size_t)(ts - t) * Hdim] * w[h * CKW + j];
  }
  out[i] = siluf(siluf(acc));
}

// row-wise L2 normalize (one 256-thread block per row)
__global__ void k_l2norm(float* __restrict__ buf) {
  __shared__ float red[256];
  const int r = blockIdx.x, t = threadIdx.x;
  float x0 = buf[(size_t)r * Hdim + t], x1 = buf[(size_t)r * Hdim + t + 256];
  red[t] = x0 * x0 + x1 * x1;
  __syncthreads();
  for (int off = 128; off > 0; off >>= 1) {
    if (t < off) red[t] += red[t + off];
    __syncthreads();
  }
  float inv = 1.f / fmaxf(sqrtf(red[0]), 1e-12f);
  buf[(size_t)r * Hdim + t]       = x0 * inv;
  buf[(size_t)r * Hdim + t + 256] = x1 * inv;
}

// ---------------------------------------------------------------------------
// Per-(batch, chunk) WY precompute: one wave per block.
//   G = (beta*K) K^T;  T = (I + strict_tril(G))^{-1}
//   Wg = T (beta*K);   Ug = T (beta*V);   Lg = tril(Q K^T, diag incl.)
// ---------------------------------------------------------------------------
__global__ void k_chunk_prep(const float* __restrict__ kc, const float* __restrict__ qc,
                             const float* __restrict__ vc, const float* __restrict__ beta,
                             float* __restrict__ Wg, float* __restrict__ Ug,
                             float* __restrict__ Lg) {
  __shared__ float sG[16 * 16];
  __shared__ float sT[16 * 17];
  __shared__ float sb[16];
  const int bc = blockIdx.x;          // b*NC + c
  const int b = bc / NC, c = bc % NC;
  const int lane = threadIdx.x & 31, hl = lane >> 4, idx = lane & 15;
  const size_t base = (size_t)b * Tlen * Hdim + (size_t)c * CH * Hdim;
  const int rb = b * Tlen + c * CH;
  if (lane < 16) sb[lane] = beta[rb + lane];
  __syncthreads();

  // G = (beta*K) K^T   and   L = Q K^T  (two wmma chains over K=512)
  {
    const float brow = sb[idx];
    v8f accG = {}, accL = {};
    for (int kk = 0; kk < Hdim / 4; ++kk) {
      const int k0 = kk * 4 + hl * 2;
      float kx = kc[base + idx * Hdim + k0], ky = kc[base + idx * Hdim + k0 + 1];
      v2f ag, aq, bb;
      ag.x = brow * kx;  ag.y = brow * ky;
      aq.x = qc[base + idx * Hdim + k0];  aq.y = qc[base + idx * Hdim + k0 + 1];
      bb.x = kx;  bb.y = ky;               // B(k,n) = K[n][k], col n == idx
      accG = __builtin_amdgcn_wmma_f32_16x16x4_f32(false, ag, false, bb, (short)0, accG,
                                                   false, false);
      accL = __builtin_amdgcn_wmma_f32_16x16x4_f32(false, aq, false, bb, (short)0, accL,
                                                   false, false);
    }
#pragma unroll
    for (int v = 0; v < 8; ++v) {
      int r = v + 8 * hl;
      sG[r * 16 + idx] = accG[v];
      Lg[(size_t)bc * 256 + r * 16 + idx] = (idx <= r) ? accL[v] : 0.f;  // incl. diag
    }
  }
  __syncthreads();

  // T = (I + strict_tril(G))^{-1} by forward substitution; lane j owns column j
  if (lane < 16) {
    for (int i = 0; i < 16; ++i) sT[i * 17 + lane] = (i == lane) ? 1.f : 0.f;
    for (int i = 1; i < 16; ++i) {
      float s = 0.f;
      for (int m = 0; m < i; ++m) s += sG[i * 16 + m] * sT[m * 17 + lane];
      sT[i * 17 + lane] -= s;
    }
  }
  __syncthreads();

  // Wg = T (beta*K),  Ug = T (beta*V)   -- K-dim = 16 (4 wmma steps) per 16-col tile
  for (int nt = 0; nt < Hdim / 16; ++nt) {
    v8f aw = {}, au = {};
#pragma unroll
    for (int kk = 0; kk < 4; ++kk) {
      const int k0 = kk * 4 + hl * 2;
      v2f a, bw, bu;
      a.x = sT[idx * 17 + k0];  a.y = sT[idx * 17 + k0 + 1];
      bw.x = sb[k0]     * kc[base + (size_t)k0 * Hdim + nt * 16 + idx];
      bw.y = sb[k0 + 1] * kc[base + (size_t)(k0 + 1) * Hdim + nt * 16 + idx];
      bu.x = sb[k0]     * vc[base + (size_t)k0 * Hdim + nt * 16 + idx];
      bu.y = sb[k0 + 1] * vc[base + (size_t)(k0 + 1) * Hdim + nt * 16 + idx];
      aw = __builtin_amdgcn_wmma_f32_16x16x4_f32(false, a, false, bw, (short)0, aw,
                                                 false, false);
      au = __builtin_amdgcn_wmma_f32_16x16x4_f32(false, a, false, bu, (short)0, au,
                                                 false, false);
    }
#pragma unroll
    for (int v = 0; v < 8; ++v) {
      int r = v + 8 * hl;
      Wg[base + (size_t)r * Hdim + nt * 16 + idx] = aw[v];
      Ug[base + (size_t)r * Hdim + nt * 16 + idx] = au[v];
    }
  }
}

// ---------------------------------------------------------------------------
// Chunked scan, async double-buffered staging. Grid (H/32, B); each block owns
// Z[:, jb:jb+32] (Z = S^T) in LDS across 32 sequential chunks. Per chunk:
//   Uq = U - W @ Z;   O = Q @ Z + L @ Uq;   Z += K^T @ Uq     (all fp32 WMMA)
// Next chunk's K/W/Q/U/L are fetched with GLOBAL_LOAD_ASYNC_TO_LDS_B128 while
// the current chunk computes; waves pay s_wait_asynccnt 0 + barrier at the end.
// Phase A splits K=512 across 2 wave-groups to halve the dependent WMMA chain.
// ---------------------------------------------------------------------------
__global__ void k_scan(const float* __restrict__ kc, const float* __restrict__ qc,
                       const float* __restrict__ Wg, const float* __restrict__ Ug,
                       const float* __restrict__ Lg, float* __restrict__ obuf,
                       float* __restrict__ Sfin) {
  extern __shared__ float ls[];
  float* sS   = ls;                         // [512][33]
  float* sKWQ = sS + 512 * 33;              // 2 bufs * {K,W,Q} * [16][SROW]
  float* sUb  = sKWQ + 2 * 3 * 16 * SROW;   // 2 * [16][32]
  float* sLb  = sUb + 2 * 512;              // 2 * [16][16]
  float* sUq  = sLb + 2 * 256;              // [16][32]
  float* sO   = sUq + 512;                  // [16][32]
  float* sPart = sO + 512;                  // 4 * [16][16] split-K partials

  const int jb = blockIdx.x * 32;
  const int b  = blockIdx.y;
  const int tid = threadIdx.x;
  const int lane = tid & 31, hl = lane >> 4, idx = lane & 15, wave = tid >> 5;

  for (int i = tid; i < 512 * 33; i += 256) sS[i] = 0.f;

  // ---- async staging of one chunk into buffer p (6336 b128 transfers) ----
  auto issue_chunk = [&](int c, int p) {
    const size_t base = (size_t)b * Tlen * Hdim + (size_t)c * CH * Hdim;
    float* dK = sKWQ + p * (3 * 16 * SROW);
    float* dW = dK + 16 * SROW;
    float* dQ = dW + 16 * SROW;
    float* dU = sUb + p * 512;
    float* dL = sLb + p * 256;
    for (int i = tid; i < 6336; i += 256) {
      const float* gp;
      float* lp;
      if (i < 6144) {
        int arr = i >> 11;                 // 0=K 1=W 2=Q
        int j = i & 2047;
        int r = j >> 7, c4 = (j & 127) << 2;
        const float* gsrc = (arr == 0) ? kc : ((arr == 1) ? Wg : qc);
        float* ldst = (arr == 0) ? dK : ((arr == 1) ? dW : dQ);
        gp = gsrc + base + (size_t)r * Hdim + c4;
        lp = ldst + r * SROW + c4;
      } else if (i < 6272) {
        int j = i - 6144;
        int r = j >> 3, c4 = (j & 7) << 2;
        gp = Ug + base + (size_t)r * Hdim + jb + c4;
        lp = dU + r * 32 + c4;
      } else {
        int j = i - 6272;
        gp = Lg + (size_t)(b * NC + c) * 256 + (j << 2);
        lp = dL + (j << 2);
      }
      async_b128(gp, lp);
    }
  };

  issue_chunk(0, 0);
  wait_async0();
  __syncthreads();

  for (int c = 0; c < NC; ++c) {
    const int p = c & 1;
    if (c + 1 < NC) issue_chunk(c + 1, p ^ 1);   // overlap with compute below

    float* sK = sKWQ + p * (3 * 16 * SROW);
    float* sW = sK + 16 * SROW;
    float* sQ = sW + 16 * SROW;
    float* sU = sUb + p * 512;
    float* sL = sLb + p * 256;

    // Phase A (8 waves): outid = wave>>1 in {Uq0,Uq1,O0,O1}, kh = wave&1 splits K.
    {
      const int outid = wave >> 1;
      const int which = outid >> 1;      // 0: W@Z (for Uq), 1: Q@Z (for O)
      const int nt    = outid & 1;
      const int kh    = wave & 1;
      const float* Am = which ? sQ : sW;
      v8f acc = {};
      for (int kk = 0; kk < 64; ++kk) {
        const int k0 = kh * 256 + kk * 4 + hl * 2;
        v2f a, bb;
        a.x = Am[idx * SROW + k0];  a.y = Am[idx * SROW + k0 + 1];
        bb.x = sS[k0 * 33 + nt * 16 + idx];
        bb.y = sS[(k0 + 1) * 33 + nt * 16 + idx];
        acc = __builtin_amdgcn_wmma_f32_16x16x4_f32(false, a, false, bb, (short)0, acc,
                                                    false, false);
      }
      if (kh == 1) {
#pragma unroll
        for (int v = 0; v < 8; ++v) sPart[outid * 256 + (v + 8 * hl) * 16 + idx] = acc[v];
      }
      __syncthreads();
      if (kh == 0) {
#pragma unroll
        for (int v = 0; v < 8; ++v) {
          int r = v + 8 * hl;
          float tot = acc[v] + sPart[outid * 256 + r * 16 + idx];
          if (which == 0) sUq[r * 32 + nt * 16 + idx] = sU[r * 32 + nt * 16 + idx] - tot;
          else            sO [r * 32 + nt * 16 + idx] = tot;
        }
      }
    }
    __syncthreads();

    // Phase B: Z += K^T @ Uq  (64 tiles of 16x16, K=16) split across 8 waves
    for (int tt = wave; tt < 64; tt += 8) {
      const int rt = tt >> 1, ct = tt & 1;
      v8f acc = {};
#pragma unroll
      for (int kk = 0; kk < 4; ++kk) {
        const int k0 = kk * 4 + hl * 2;
        v2f a, bb;
        a.x = sK[k0 * SROW + rt * 16 + idx];
        a.y = sK[(k0 + 1) * SROW + rt * 16 + idx];
        bb.x = sUq[k0 * 32 + ct * 16 + idx];
        bb.y = sUq[(k0 + 1) * 32 + ct * 16 + idx];
        acc = __builtin_amdgcn_wmma_f32_16x16x4_f32(false, a, false, bb, (short)0, acc,
                                                    false, false);
      }
#pragma unroll
      for (int v = 0; v < 8; ++v)
        sS[(rt * 16 + v + 8 * hl) * 33 + ct * 16 + idx] += acc[v];
    }
    // waves 0-1: O = O1 + L @ Uq, write to global
    if (wave < 2) {
      const int ct = wave;
      v8f acc = {};
#pragma unroll
      for (int kk = 0; kk < 4; ++kk) {
        const int k0 = kk * 4 + hl * 2;
        v2f a, bb;
        a.x = sL[idx * 16 + k0];  a.y = sL[idx * 16 + k0 + 1];
        bb.x = sUq[k0 * 32 + ct * 16 + idx];
        bb.y = sUq[(k0 + 1) * 32 + ct * 16 + idx];
        acc = __builtin_amdgcn_wmma_f32_16x16x4_f32(false, a, false, bb, (short)0, acc,
                                                    false, false);
      }
#pragma unroll
      for (int v = 0; v < 8; ++v) {
        int r = v + 8 * hl;
        obuf[(size_t)b * Tlen * Hdim + (size_t)(c * CH + r) * Hdim + jb + ct * 16 + idx] =
            sO[r * 32 + ct * 16 + idx] + acc[v];
      }
    }
    wait_async0();        // next chunk's staged tiles resident
    __syncthreads();
  }
  // S_final[b][j][i] = Z[i][j]
  for (int i = tid; i < 512 * 32; i += 256) {
    int row = i >> 5, jj = i & 31;
    Sfin[(size_t)b * Hdim * Hdim + (size_t)(jb + jj) * Hdim + row] = sS[row * 33 + jj];
  }
}

// RMSNorm with gain, in place (one block per row)
__global__ void k_rmsnorm(float* __restrict__ o, const float* __restrict__ g) {
  __shared__ float red[256];
  const int r = blockIdx.x, t = threadIdx.x;
  float x0 = o[(size_t)r * Hdim + t], x1 = o[(size_t)r * Hdim + t + 256];
  red[t] = x0 * x0 + x1 * x1;
  __syncthreads();
  for (int off = 128; off > 0; off >>= 1) {
    if (t < off) red[t] += red[t + off];
    __syncthreads();
  }
  float rms = sqrtf(red[0] / (float)Hdim + 1e-5f);
  o[(size_t)r * Hdim + t]       = x0 / rms * g[t];
  o[(size_t)r * Hdim + t + 256] = x1 / rms * g[t + 256];
}

extern "C" void kernel_launch(void* const* d_in, const int* in_sizes, int n_in,
                              void* d_out, int out_size, void* d_ws, size_t ws_size,
                              hipStream_t stream) {
  (void)in_sizes; (void)n_in; (void)out_size; (void)ws_size;
  const float* x     = (const float*)d_in[0];
  const float* Wk    = (const float*)d_in[1];
  const float* Wq    = (const float*)d_in[2];
  const float* Wv    = (const float*)d_in[3];
  const float* convk = (const float*)d_in[4];
  const float* convq = (const float*)d_in[5];
  const float* convv = (const float*)d_in[6];
  const float* Wb    = (const float*)d_in[7];
  const float* g     = (const float*)d_in[8];
  const float* Wo    = (const float*)d_in[9];
  float* out = (float*)d_out;
  float* ws  = (float*)d_ws;

  // workspace map (buffers reused once dead): ~24 MB total
  float* kraw = ws;                     // -> reused as Wg (k_cumdecay)
  float* qraw = ws + (size_t)BTH;       // -> reused as Ug
  float* vraw = ws + 2 * (size_t)BTH;   // -> reused as obuf
  float* kc   = ws + 3 * (size_t)BTH;
  float* qc   = ws + 4 * (size_t)BTH;
  float* vc   = ws + 5 * (size_t)BTH;
  float* beta = ws + 6 * (size_t)BTH;
  float* Lg   = beta + 4096;

  dim3 gGemm(BT / 16, Hdim / 16);
  k_gemm_xWT<<<gGemm, 32, 0, stream>>>(x, Wk, kraw);
  k_gemm_xWT<<<gGemm, 32, 0, stream>>>(x, Wq, qraw);
  k_gemm_xWT<<<gGemm, 32, 0, stream>>>(x, Wv, vraw);
  k_beta<<<(BT + 255) / 256, 256, 0, stream>>>(x, Wb, beta);

  const int nblk = (BTH + 255) / 256;
  k_conv_silu<<<nblk, 256, 0, stream>>>(kraw, convk, kc);
  k_conv_silu<<<nblk, 256, 0, stream>>>(qraw, convq, qc);
  k_conv_silu<<<nblk, 256, 0, stream>>>(vraw, convv, vc);
  k_l2norm<<<BT, 256, 0, stream>>>(kc);
  k_l2norm<<<BT, 256, 0, stream>>>(qc);

  k_chunk_prep<<<Bsz * NC, 32, 0, stream>>>(kc, qc, vc, beta, /*Wg*/ kraw,
                                            /*Ug*/ qraw, Lg);

  size_t shmem = (size_t)(512 * 33 + 2 * 3 * 16 * SROW + 2 * 512 + 2 * 256 +
                          512 + 512 + 1024) * sizeof(float);
  k_scan<<<dim3(Hdim / 32, Bsz), 256, shmem, stream>>>(kc, qc, /*Wg*/ kraw, /*Ug*/ qraw,
                                                       Lg, /*obuf*/ vraw,
                                                       /*S_final*/ out + (size_t)BTH);

  k_rmsnorm<<<BT, 256, 0, stream>>>(vraw, g);
  k_gemm_xWT<<<gGemm, 32, 0, stream>>>(vraw, Wo, out);
}